// Word_Graph_Attention_71511205478705
// MI455X (gfx1250) — compile-verified
//
#include <hip/hip_runtime.h>
#include <math.h>

// Shapes from reference
#define Bsz 32
#define Ssz 256
#define Nn  50
#define Dd  100
#define QDd 768

typedef float v2f __attribute__((ext_vector_type(2)));
typedef float v8f __attribute__((ext_vector_type(8)));

// ---------------- Kernel 0: Q projection + L2 normalize -> Qn[b][e] ----------
__global__ __launch_bounds__(128) void qproj_kernel(
    const float* __restrict__ q,   // [B, QD]
    const float* __restrict__ Qw,  // [D, QD]
    const float* __restrict__ Qb,  // [D]
    float* __restrict__ Qn)        // [B, D] (workspace)
{
    const int b = blockIdx.x;
    const int t = threadIdx.x;          // 128 threads
    __shared__ float sq[QDd];
    __shared__ float sred[128];

    for (int i = t; i < QDd; i += 128) sq[i] = q[b * QDd + i];
    __syncthreads();

    float dot = 0.f;
    if (t < Dd) {
        const float* wr = Qw + (size_t)t * QDd;
        float acc = Qb[t];
        for (int k = 0; k < QDd; ++k) acc = fmaf(sq[k], wr[k], acc);
        dot = acc;
    }
    sred[t] = (t < Dd) ? dot * dot : 0.f;
    __syncthreads();
    for (int off = 64; off > 0; off >>= 1) {
        if (t < off) sred[t] += sred[t + off];
        __syncthreads();
    }
    const float nrm  = sqrtf(sred[0]);
    const float rinv = 1.0f / fmaxf(nrm, 1e-12f);
    if (t < Dd) Qn[b * Dd + t] = dot * rinv;
}

// ---------------- Kernel 1: fused projection + attention per (b,s) -----------
// Persistent block handles TPB consecutive (b,s) tiles; Vw^T/Vb/Qn staged once.
// K_tile = v[b,s] @ Vw^T + Vb via V_WMMA_F32_16X16X4_F32 (no guards in k-loop:
// 7th N-tile overlaps at column origin 84, duplicates masked in the epilogue),
// then cosine scores vs Qn[b], leaky-softmax over n, weighted sum -> out.
#define NT    7
#define VST   101    // sV row stride (A reads: 37j mod 64, conflict-free)
#define BST   104    // sB row stride (cols 0..103; max col read = 99)
#define TPB   4      // (b,s) tiles per persistent block; 256 % TPB == 0

// per-tile B column origins: 6 regular tiles + overlapping tail tile at 84
__constant__ int kColOrg[NT] = {0, 16, 32, 48, 64, 80, 84};

__global__ __launch_bounds__(128) void wga_main_kernel(
    const float* __restrict__ v,    // [B,S,N,D]
    const float* __restrict__ Vw,   // [D,D]
    const float* __restrict__ Vb,   // [D]
    const float* __restrict__ Qn,   // [B,D]
    float* __restrict__ out)        // [B,S,D]
{
    __shared__ float sV[Nn * VST];       // v tile, rows 0..49
    __shared__ float sB[Dd * BST];       // Vw^T : sB[d*BST + e]
    __shared__ float sQn[112];
    __shared__ float sVb[112];
    __shared__ float sOut[112];
    __shared__ float sScore[64];
    __shared__ float sAtt[64];

    const int bs0  = blockIdx.x * TPB;   // first tile; b is block-invariant
    const int b    = bs0 / Ssz;
    const int tid  = threadIdx.x;        // 128 = 4 waves
    const int lane = tid & 31;
    const int wave = tid >> 5;
    const int half = lane >> 4;          // K-half selector for f32 WMMA layout
    const int l16  = lane & 15;

    // ---- one-time staging: Vw transposed, bias, normalized query ----
    for (int i = tid; i < (Dd * Dd) / 4; i += 128) {
        float4 x = ((const float4*)Vw)[i];
        int f = i * 4;
        int e = f / Dd, d = f - e * Dd;
        sB[(d + 0) * BST + e] = x.x;
        sB[(d + 1) * BST + e] = x.y;
        sB[(d + 2) * BST + e] = x.z;
        sB[(d + 3) * BST + e] = x.w;
    }
    for (int i = tid; i < 112; i += 128) {
        sQn[i] = (i < Dd) ? Qn[b * Dd + i] : 0.f;
        sVb[i] = (i < Dd) ? Vb[i] : 0.f;
    }

    const int   m0   = wave * 16;
    const int   arow = min(m0 + l16, Nn - 1);   // rows >=50: garbage, discarded
    const float m6   = (l16 >= 12) ? 1.f : 0.f; // tail-tile unique cols: 96..99
    const int   c6   = 84 + l16;

    for (int it = 0; it < TPB; ++it) {
        const int bs = bs0 + it;

        // ---- stage v tile (50x100, coalesced B128 loads) + zero sOut ----
        const float* vt = v + (size_t)bs * (Nn * Dd);
        for (int i = tid; i < (Nn * Dd) / 4; i += 128) {
            float4 x = ((const float4*)vt)[i];
            int f = i * 4;
            int n = f / Dd, d = f - n * Dd;   // D % 4 == 0: never crosses a row
            float* p = &sV[n * VST + d];
            p[0] = x.x; p[1] = x.y; p[2] = x.z; p[3] = x.w;
        }
        for (int i = tid; i < 112; i += 128) sOut[i] = 0.f;
        __syncthreads();

        // ---- WMMA f32 16x16x4: wave owns rows [m0..m0+15], 7 N-tiles ----
        v8f acc[NT];
#pragma unroll
        for (int t = 0; t < NT; ++t)
            acc[t] = (v8f){0.f,0.f,0.f,0.f,0.f,0.f,0.f,0.f};

        const float* aptr = &sV[arow * VST + 2 * half];
#pragma unroll 5
        for (int kk = 0; kk < Dd; kk += 4) {
            v2f a;
            a.x = aptr[kk];
            a.y = aptr[kk + 1];
            const float* bbase = &sB[(kk + 2 * half) * BST + l16];
#pragma unroll
            for (int t = 0; t < NT; ++t) {
                v2f bf;
                bf.x = bbase[kColOrg[t]];
                bf.y = bbase[kColOrg[t] + BST];
                acc[t] = __builtin_amdgcn_wmma_f32_16x16x4_f32(
                    false, a, false, bf, (short)0, acc[t], false, false);
            }
        }

        // ---- bias add + per-row ||K||^2 and Qn.K partials ----
        float ss[8], pp[8];
#pragma unroll
        for (int r = 0; r < 8; ++r) { ss[r] = 0.f; pp[r] = 0.f; }
#pragma unroll
        for (int t = 0; t < 6; ++t) {
            const float bias = sVb[t * 16 + l16];
            const float qv   = sQn[t * 16 + l16];
#pragma unroll
            for (int r = 0; r < 8; ++r) {
                float kv = acc[t][r] + bias;
                acc[t][r] = kv;
                ss[r] = fmaf(kv, kv, ss[r]);
                pp[r] = fmaf(kv, qv, pp[r]);
            }
        }
        {   // overlapping tail tile: only cols 96..99 (l16>=12) are unique
            const float bias = sVb[c6];
            const float qv   = m6 * sQn[c6];
#pragma unroll
            for (int r = 0; r < 8; ++r) {
                float kv = acc[6][r] + bias;
                acc[6][r] = kv;
                ss[r] = fmaf(m6 * kv, kv, ss[r]);
                pp[r] = fmaf(kv, qv, pp[r]);
            }
        }
#pragma unroll
        for (int m = 1; m < 16; m <<= 1) {
#pragma unroll
            for (int r = 0; r < 8; ++r) {
                ss[r] += __shfl_xor(ss[r], m, 32);
                pp[r] += __shfl_xor(pp[r], m, 32);
            }
        }
        if (l16 == 0) {
#pragma unroll
            for (int r = 0; r < 8; ++r) {
                const int row = m0 + half * 8 + r;      // 0..63
                float l2 = sqrtf(ss[r]);
                l2 = (l2 == 0.f) ? 1e-6f : l2;
                float sc = pp[r] / l2;                  // cosine score
                sc = (sc == 0.f) ? -10000.f : sc;       // zero-score mask
                sc = (sc > 0.f) ? sc : 0.01f * sc;      // leaky relu
                sScore[row] = sc;
            }
        }
        __syncthreads();

        // ---- softmax over n (50), wave 0 ----
        if (wave == 0) {
            const int n0 = lane, n1 = lane + 32;
            float s0 = sScore[n0];                       // n0 < 50 always? no: <32
            float s1 = (n1 < Nn) ? sScore[n1] : -1e30f;
            float mx = fmaxf(s0, s1);
#pragma unroll
            for (int m = 1; m < 32; m <<= 1) mx = fmaxf(mx, __shfl_xor(mx, m, 32));
            float e0 = expf(s0 - mx);
            float e1 = (n1 < Nn) ? expf(s1 - mx) : 0.f;
            float sum = e0 + e1;
#pragma unroll
            for (int m = 1; m < 32; m <<= 1) sum += __shfl_xor(sum, m, 32);
            float a0 = e0 / sum;
            float a1 = e1 / sum;
            if (a0 == 1.0f / (float)Nn) a0 = 0.f;        // uniform-row zeroing
            if (a1 == 1.0f / (float)Nn) a1 = 0.f;
            sAtt[n0] = a0;
            sAtt[n1] = (n1 < Nn) ? a1 : 0.f;
        }
        __syncthreads();

        // ---- out[b,s,e] = sum_n att[n] * K[n,e] ----
        float w[8];
#pragma unroll
        for (int r = 0; r < 8; ++r) w[r] = sAtt[m0 + half * 8 + r]; // 0 if padded
#pragma unroll
        for (int t = 0; t < 6; ++t) {
            float o = 0.f;
#pragma unroll
            for (int r = 0; r < 8; ++r) o = fmaf(w[r], acc[t][r], o);
            o += __shfl_xor(o, 16, 32);          // fold row-halves (same column)
            if (half == 0)
                atomicAdd(&sOut[t * 16 + l16], o);   // col <= 95, always valid
        }
        {   // tail tile: contribute only unique cols 96..99
            float o = 0.f;
#pragma unroll
            for (int r = 0; r < 8; ++r) o = fmaf(w[r], acc[6][r], o);
            o += __shfl_xor(o, 16, 32);
            if (half == 0 && l16 >= 12)
                atomicAdd(&sOut[c6], o);
        }
        __syncthreads();
        for (int i = tid; i < Dd; i += 128)
            out[(size_t)bs * Dd + i] = sOut[i];
        __syncthreads();   // protect sV/sOut before next tile's staging
    }
}

// ---------------- launch -----------------------------------------------------
extern "C" void kernel_launch(void* const* d_in, const int* in_sizes, int n_in,
                              void* d_out, int out_size, void* d_ws, size_t ws_size,
                              hipStream_t stream) {
    // setup_inputs order: input_ent, q, k, v, Q_w, Q_b, V_w, V_b
    const float* q  = (const float*)d_in[1];
    const float* v  = (const float*)d_in[3];
    const float* Qw = (const float*)d_in[4];
    const float* Qb = (const float*)d_in[5];
    const float* Vw = (const float*)d_in[6];
    const float* Vb = (const float*)d_in[7];
    float* out = (float*)d_out;
    float* Qn  = (float*)d_ws;                    // [B, D] scratch

    qproj_kernel<<<Bsz, 128, 0, stream>>>(q, Qw, Qb, Qn);
    wga_main_kernel<<<(Bsz * Ssz) / TPB, 128, 0, stream>>>(v, Vw, Vb, Qn, out);
}